// E3EqMulHeadAttBlock_62234076119440
// MI455X (gfx1250) — compile-verified
//
#include <hip/hip_runtime.h>
#include <hip/hip_bf16.h>

// ---------------------------------------------------------------------------
// E3EqMulHeadAttBlock on gfx1250 — fp32 WMMA (V_WMMA_F32_16X16X4_F32) pipeline
// Round 4: round-3 software pipelining + __launch_bounds__ to raise the VGPR
// allocation cap (round 3 spilled all fragment buffers to scratch at the
// default 8-waves/SIMD budget).
//
// ws (floats): Qs[0..]         8388608   (head-split: (a*8+h)*8192 + rp*16+fr)
//              Ks[8388608..]   8388608   (same layout)
//              Vt[16777216..]  8388608   (transposed: (cl*128+f)*128 + b)
//              dp[25165824..]  131072    ((a*8+h)*128 + c)
//              WqT/WkT/WvT     3x16384   at 25296896
//              WoT             131072    at 25346048   (g*1024 + k)
//              WtaT/WtbT/WtoT  3x16384   at 25477120
//              out1 overlays Qs.
// ---------------------------------------------------------------------------

typedef float v2f __attribute__((ext_vector_type(2)));
typedef float v8f __attribute__((ext_vector_type(8)));

__device__ __forceinline__ v8f wmma4(v2f a, v2f b, v8f c) {
  return __builtin_amdgcn_wmma_f32_16x16x4_f32(false, a, false, b,
                                               (short)0, c, false, false);
}
__device__ __forceinline__ v8f zero8() {
  v8f z;
#pragma unroll
  for (int i = 0; i < 8; ++i) z[i] = 0.0f;
  return z;
}
__device__ __forceinline__ v2f ld2(const float* p) { return *(const v2f*)p; }

// ---------------------------------------------------------------------------
// K0: generic transpose  dst[n*K + k] = src[k*N + n]
// ---------------------------------------------------------------------------
__global__ void k_transpose(const float* __restrict__ src, float* __restrict__ dst,
                            int K, int N) {
  const int i = blockIdx.x * 256 + threadIdx.x;
  if (i < K * N) {
    const int k = i / N, n = i - k * N;
    dst[n * K + k] = src[i];
  }
}

// ---------------------------------------------------------------------------
// K1: Q/K/V = x @ W — one wave per 16-row tile; double-buffered fragments.
// ~115 VGPRs needed -> allow up to 256 (min 4 waves/EU).
// ---------------------------------------------------------------------------
__global__ void __launch_bounds__(32, 4)
k_qkv(const float* __restrict__ x,  const float* __restrict__ WqT,
      const float* __restrict__ WkT, const float* __restrict__ WvT,
      float* __restrict__ Qs, float* __restrict__ Ks, float* __restrict__ Vt) {
  const int lane = threadIdx.x & 31, half = lane >> 4, mr = lane & 15;
  const int rt = blockIdx.x, z = blockIdx.y;          // z: 0=Q 1=K 2=V
  const float* W = (z == 0) ? WqT : ((z == 1) ? WkT : WvT);
  const int row0 = rt * 16;
  const float* xp = x + (long)(row0 + mr) * 128;
  const float* wb = W + (long)mr * 128;               // + t*2048 + k0
  v8f acc[8];
#pragma unroll
  for (int t = 0; t < 8; ++t) acc[t] = zero8();

  v2f a0 = ld2(xp + half * 2);
  v2f b0[8];
#pragma unroll
  for (int t = 0; t < 8; ++t) b0[t] = ld2(wb + t * 2048 + half * 2);
  for (int kk = 1; kk < 32; ++kk) {
    const int k1 = kk * 4 + half * 2;
    const v2f a1 = ld2(xp + k1);
    v2f b1[8];
#pragma unroll
    for (int t = 0; t < 8; ++t) b1[t] = ld2(wb + t * 2048 + k1);
#pragma unroll
    for (int t = 0; t < 8; ++t) acc[t] = wmma4(a0, b0[t], acc[t]);
    a0 = a1;
#pragma unroll
    for (int t = 0; t < 8; ++t) b0[t] = b1[t];
  }
#pragma unroll
  for (int t = 0; t < 8; ++t) acc[t] = wmma4(a0, b0[t], acc[t]);

  if (z < 2) {
    float* O = (z == 0) ? Qs : Ks;
#pragma unroll
    for (int t = 0; t < 8; ++t)
#pragma unroll
      for (int v = 0; v < 8; ++v) {
        const int row = row0 + v + half * 8;          // (a, rp)
        const int f   = t * 16 + mr;                  // fr*8 + h
        O[(long)((row >> 9) * 8 + (f & 7)) * 8192 +
          ((row & 511) << 4) + (f >> 3)] = acc[t][v];
      }
  } else {
#pragma unroll
    for (int t = 0; t < 8; ++t)
#pragma unroll
      for (int v = 0; v < 8; ++v) {
        const int row = row0 + v + half * 8;          // (b, cl)
        const int f   = t * 16 + mr;
        Vt[((long)(row & 511) * 128 + f) * 128 + (row >> 9)] = acc[t][v];
      }
  }
}

// ---------------------------------------------------------------------------
// K2: dp[a,c,h] — 4 independent accumulator chains + double-buffered loads.
// ~60 VGPRs -> cap 256 is ample.
// ---------------------------------------------------------------------------
__global__ void __launch_bounds__(32, 4)
k_dp(const float* __restrict__ Qs, const float* __restrict__ Ks,
     const float* __restrict__ mask, float* __restrict__ dp) {
  const int lane = threadIdx.x & 31, half = lane >> 4, mr = lane & 15;
  const int at = blockIdx.x, ct = blockIdx.y, h = blockIdx.z;
  const float* qp = Qs + (long)((at * 16 + mr) * 8 + h) * 8192 + half * 2;
  const float* kp = Ks + (long)((ct * 16 + mr) * 8 + h) * 8192 + half * 2;
  v8f acc[4];
#pragma unroll
  for (int u = 0; u < 4; ++u) acc[u] = zero8();

  v2f qa[4], kb[4];
#pragma unroll
  for (int u = 0; u < 4; ++u) { qa[u] = ld2(qp + u * 4); kb[u] = ld2(kp + u * 4); }
  for (int kk = 4; kk < 2048; kk += 4) {
    v2f qn[4], kn[4];
#pragma unroll
    for (int u = 0; u < 4; ++u) {
      qn[u] = ld2(qp + (kk + u) * 4);
      kn[u] = ld2(kp + (kk + u) * 4);
    }
#pragma unroll
    for (int u = 0; u < 4; ++u) acc[u] = wmma4(qa[u], kb[u], acc[u]);
#pragma unroll
    for (int u = 0; u < 4; ++u) { qa[u] = qn[u]; kb[u] = kn[u]; }
  }
#pragma unroll
  for (int u = 0; u < 4; ++u) acc[u] = wmma4(qa[u], kb[u], acc[u]);
  const v8f accs = (acc[0] + acc[1]) + (acc[2] + acc[3]);

#pragma unroll
  for (int v = 0; v < 8; ++v) {
    const int arow = at * 16 + v + half * 8;
    const int ccol = ct * 16 + mr;
    dp[(arow * 8 + h) * 128 + ccol] = accs[v] * 0.125f + mask[arow * 128 + ccol];
  }
}

// ---------------------------------------------------------------------------
// K3: softmax over c for each (a,h) row of 128 — one wave per row.
// ---------------------------------------------------------------------------
__global__ void k_softmax(float* __restrict__ dp) {
  const int lane = threadIdx.x & 31;
  float* row = dp + (long)blockIdx.x * 128;
  float v0 = row[lane], v1 = row[lane + 32], v2 = row[lane + 64], v3 = row[lane + 96];
  float m = fmaxf(fmaxf(v0, v1), fmaxf(v2, v3));
#pragma unroll
  for (int s = 16; s >= 1; s >>= 1) m = fmaxf(m, __shfl_xor(m, s, 32));
  v0 = __expf(v0 - m); v1 = __expf(v1 - m); v2 = __expf(v2 - m); v3 = __expf(v3 - m);
  float sum = v0 + v1 + v2 + v3;
#pragma unroll
  for (int s = 16; s >= 1; s >>= 1) sum += __shfl_xor(sum, s, 32);
  const float inv = 1.0f / sum;
  row[lane] = v0 * inv; row[lane + 32] = v1 * inv;
  row[lane + 64] = v2 * inv; row[lane + 96] = v3 * inv;
}

// ---------------------------------------------------------------------------
// K4: out1[(a,cl),g] = x + sum_{f,h} (sum_b w[a,b,h] V[b,cl,f]) Wo[f*8+h, g]
// Both matmul stages double-buffered. ~175 VGPRs -> allow 512 (min 2 waves).
// ---------------------------------------------------------------------------
__global__ void __launch_bounds__(32, 2)
k_att(const float* __restrict__ w, const float* __restrict__ Vt,
      const float* __restrict__ WoT, const float* __restrict__ x,
      float* __restrict__ out1) {
  __shared__ float lds[16 * 128];
  const int lane = threadIdx.x & 31, half = lane >> 4, mr = lane & 15;
  const int at = blockIdx.x;            // a-tile 0..7
  const int cl = blockIdx.y;            // (c*4+l) 0..511
  const float* wp = w + (long)(at * 16 + mr) * 1024;   // + h*128 + b0
  v8f oacc[8];
#pragma unroll
  for (int t = 0; t < 8; ++t) oacc[t] = zero8();

  for (int fs = 0; fs < 8; ++fs) {      // f-subtile of 16
    // ---- stage 1: att_h = w_h @ V  (contraction over b, 8 head chains) ----
    v8f aacc[8];
#pragma unroll
    for (int h = 0; h < 8; ++h) aacc[h] = zero8();
    const float* vp = Vt + (long)(cl * 128 + fs * 16 + mr) * 128;
    v2f bf0 = ld2(vp + half * 2);
    v2f af0[8];
#pragma unroll
    for (int h = 0; h < 8; ++h) af0[h] = ld2(wp + h * 128 + half * 2);
    for (int kk = 1; kk < 32; ++kk) {
      const int b1 = kk * 4 + half * 2;
      const v2f bf1 = ld2(vp + b1);
      v2f af1[8];
#pragma unroll
      for (int h = 0; h < 8; ++h) af1[h] = ld2(wp + h * 128 + b1);
#pragma unroll
      for (int h = 0; h < 8; ++h) aacc[h] = wmma4(af0[h], bf0, aacc[h]);
      bf0 = bf1;
#pragma unroll
      for (int h = 0; h < 8; ++h) af0[h] = af1[h];
    }
#pragma unroll
    for (int h = 0; h < 8; ++h) aacc[h] = wmma4(af0[h], bf0, aacc[h]);

    // ---- relayout: D-frag -> LDS row m, column fl*8+h, b64-packed over (h,h+1) ----
#pragma unroll
    for (int h = 0; h < 8; h += 2)
#pragma unroll
      for (int v = 0; v < 8; ++v) {
        v2f p; p.x = aacc[h][v]; p.y = aacc[h + 1][v];
        *(v2f*)(&lds[(v + half * 8) * 128 + mr * 8 + h]) = p;
      }
    __syncthreads();

    // ---- stage 2: oacc += att_sub @ Wo_sub (contraction 128, 8 chains) ----
    const float* op = WoT + (long)mr * 1024 + fs * 128;   // + t*16384 + k0
    v2f lf0 = ld2(&lds[mr * 128 + half * 2]);
    v2f bo0[8];
#pragma unroll
    for (int t = 0; t < 8; ++t) bo0[t] = ld2(op + t * 16384 + half * 2);
    for (int kk = 1; kk < 32; ++kk) {
      const int k1 = kk * 4 + half * 2;
      const v2f lf1 = ld2(&lds[mr * 128 + k1]);
      v2f bo1[8];
#pragma unroll
      for (int t = 0; t < 8; ++t) bo1[t] = ld2(op + t * 16384 + k1);
#pragma unroll
      for (int t = 0; t < 8; ++t) oacc[t] = wmma4(lf0, bo0[t], oacc[t]);
      lf0 = lf1;
#pragma unroll
      for (int t = 0; t < 8; ++t) bo0[t] = bo1[t];
    }
#pragma unroll
    for (int t = 0; t < 8; ++t) oacc[t] = wmma4(lf0, bo0[t], oacc[t]);
    __syncthreads();
  }
#pragma unroll
  for (int t = 0; t < 8; ++t)
#pragma unroll
    for (int v = 0; v < 8; ++v) {
      const long r = (long)(at * 16 + v + half * 8) * 512 + cl;
      out1[r * 128 + t * 16 + mr] = oacc[t][v] + x[r * 128 + t * 16 + mr];
    }
}

// ---------------------------------------------------------------------------
// K5: tensor_dense + gated relu. Stage 1 clusters 17 loads -> 16 WMMAs
// (128 accum VGPRs live); per-lane L-mix; stage 2 double-buffered.
// ~180 VGPRs -> allow 512 (min 2 waves).
// ---------------------------------------------------------------------------
__global__ void __launch_bounds__(32, 2)
k_tensor(const float* __restrict__ in, const float* __restrict__ WtaT,
         const float* __restrict__ WtbT, const float* __restrict__ WtoT,
         float* __restrict__ out) {
  __shared__ float lds[16 * 128];
  const int lane = threadIdx.x & 31, half = lane >> 4, mr = lane & 15;
  const int row0 = blockIdx.x * 16;
  const float* ip = in + (long)(row0 + mr) * 128;
  const float* wa = WtaT + (long)mr * 128;
  const float* wb = WtbT + (long)mr * 128;
  v8f aacc[8], bacc[8];
#pragma unroll
  for (int t = 0; t < 8; ++t) { aacc[t] = zero8(); bacc[t] = zero8(); }
  for (int kk = 0; kk < 32; ++kk) {
    const int k0 = kk * 4 + half * 2;
    const v2f af = ld2(ip + k0);
    v2f ba[8], bb[8];
#pragma unroll
    for (int t = 0; t < 8; ++t) ba[t] = ld2(wa + t * 2048 + k0);
#pragma unroll
    for (int t = 0; t < 8; ++t) bb[t] = ld2(wb + t * 2048 + k0);
#pragma unroll
    for (int t = 0; t < 8; ++t) aacc[t] = wmma4(af, ba[t], aacc[t]);
#pragma unroll
    for (int t = 0; t < 8; ++t) bacc[t] = wmma4(af, bb[t], bacc[t]);
  }
  const float is3 = 0.57735026918962576451f;  // 1/sqrt(3)
#pragma unroll
  for (int t = 0; t < 8; ++t) {
    v8f y;
    y[0] = aacc[t][0] * bacc[t][0] +
           (aacc[t][1] * bacc[t][1] + aacc[t][2] * bacc[t][2] + aacc[t][3] * bacc[t][3]) * is3;
    y[1] = aacc[t][0] * bacc[t][1] + aacc[t][1] * bacc[t][0];
    y[2] = aacc[t][0] * bacc[t][2] + aacc[t][2] * bacc[t][0];
    y[3] = aacc[t][0] * bacc[t][3] + aacc[t][3] * bacc[t][0];
    y[4] = aacc[t][4] * bacc[t][4] +
           (aacc[t][5] * bacc[t][5] + aacc[t][6] * bacc[t][6] + aacc[t][7] * bacc[t][7]) * is3;
    y[5] = aacc[t][4] * bacc[t][5] + aacc[t][5] * bacc[t][4];
    y[6] = aacc[t][4] * bacc[t][6] + aacc[t][6] * bacc[t][4];
    y[7] = aacc[t][4] * bacc[t][7] + aacc[t][7] * bacc[t][4];
#pragma unroll
    for (int v = 0; v < 8; ++v)
      lds[(v + half * 8) * 128 + t * 16 + mr] = y[v];   // A row-major [m][k]
  }
  __syncthreads();
  v8f oacc[8];
#pragma unroll
  for (int t = 0; t < 8; ++t) oacc[t] = zero8();
  const float* wo = WtoT + (long)mr * 128;
  v2f lf0 = ld2(&lds[mr * 128 + half * 2]);
  v2f bo0[8];
#pragma unroll
  for (int t = 0; t < 8; ++t) bo0[t] = ld2(wo + t * 2048 + half * 2);
  for (int kk = 1; kk < 32; ++kk) {
    const int k1 = kk * 4 + half * 2;
    const v2f lf1 = ld2(&lds[mr * 128 + k1]);
    v2f bo1[8];
#pragma unroll
    for (int t = 0; t < 8; ++t) bo1[t] = ld2(wo + t * 2048 + k1);
#pragma unroll
    for (int t = 0; t < 8; ++t) oacc[t] = wmma4(lf0, bo0[t], oacc[t]);
    lf0 = lf1;
#pragma unroll
    for (int t = 0; t < 8; ++t) bo0[t] = bo1[t];
  }
#pragma unroll
  for (int t = 0; t < 8; ++t) oacc[t] = wmma4(lf0, bo0[t], oacc[t]);

#pragma unroll
  for (int t = 0; t < 8; ++t) {
    const float g0 = (oacc[t][0] > 0.0f) ? 1.0f : 0.0f;   // gate = l==0 row
    const float g1 = (oacc[t][4] > 0.0f) ? 1.0f : 0.0f;
    oacc[t][0] *= g0; oacc[t][1] *= g0; oacc[t][2] *= g0; oacc[t][3] *= g0;
    oacc[t][4] *= g1; oacc[t][5] *= g1; oacc[t][6] *= g1; oacc[t][7] *= g1;
#pragma unroll
    for (int v = 0; v < 8; ++v)
      out[(long)(row0 + v + half * 8) * 128 + t * 16 + mr] = oacc[t][v];
  }
}

// ---------------------------------------------------------------------------
extern "C" void kernel_launch(void* const* d_in, const int* in_sizes, int n_in,
                              void* d_out, int out_size, void* d_ws, size_t ws_size,
                              hipStream_t stream) {
  const float* x    = (const float*)d_in[0];
  const float* mask = (const float*)d_in[1];
  const float* Wq   = (const float*)d_in[2];
  const float* Wk   = (const float*)d_in[3];
  const float* Wv   = (const float*)d_in[4];
  const float* Wo   = (const float*)d_in[5];
  const float* Wta  = (const float*)d_in[6];
  const float* Wtb  = (const float*)d_in[7];
  const float* Wto  = (const float*)d_in[8];

  float* ws   = (float*)d_ws;
  float* Qs   = ws;                  // 8388608
  float* Ks   = ws + 8388608;        // 8388608
  float* Vt   = ws + 16777216;       // 8388608
  float* dp   = ws + 25165824;       // 131072
  float* WqT  = ws + 25296896;       // 16384
  float* WkT  = ws + 25313280;
  float* WvT  = ws + 25329664;
  float* WoT  = ws + 25346048;       // 131072 (g*1024 + k)
  float* WtaT = ws + 25477120;
  float* WtbT = ws + 25493504;
  float* WtoT = ws + 25509888;
  float* out1 = Qs;                  // Qs dead after k_dp -> overlay

  k_transpose<<<64,  256, 0, stream>>>(Wq,  WqT,  128,  128);
  k_transpose<<<64,  256, 0, stream>>>(Wk,  WkT,  128,  128);
  k_transpose<<<64,  256, 0, stream>>>(Wv,  WvT,  128,  128);
  k_transpose<<<512, 256, 0, stream>>>(Wo,  WoT,  1024, 128);
  k_transpose<<<64,  256, 0, stream>>>(Wta, WtaT, 128,  128);
  k_transpose<<<64,  256, 0, stream>>>(Wtb, WtbT, 128,  128);
  k_transpose<<<64,  256, 0, stream>>>(Wto, WtoT, 128,  128);

  k_qkv    <<<dim3(4096, 3), 32, 0, stream>>>(x, WqT, WkT, WvT, Qs, Ks, Vt);
  k_dp     <<<dim3(8, 8, 8),  32, 0, stream>>>(Qs, Ks, mask, dp);
  k_softmax<<<1024,           32, 0, stream>>>(dp);
  k_att    <<<dim3(8, 512),   32, 0, stream>>>(dp, Vt, WoT, x, out1);
  k_tensor <<<4096,           32, 0, stream>>>(out1, WtaT, WtbT, WtoT, (float*)d_out);
}